// CrossFrameAbsoluteAttn_60790967108149
// MI455X (gfx1250) — compile-verified
//
#include <hip/hip_runtime.h>
#include <hip/hip_bf16.h>
#include <math.h>

typedef float v2f __attribute__((ext_vector_type(2)));
typedef float v8f __attribute__((ext_vector_type(8)));

#define NHEADS 9
#define EMBED  32
#define HD     (NHEADS * EMBED)   // 288
#define HW     4096               // 64*64
#define BATCH  2

// ---------------------------------------------------------------------------
// Bilinear upsample 32x32 -> 64x64, half-pixel centers, edge clamp
// (jax.image.resize 'linear', antialias=False)
// ---------------------------------------------------------------------------
__global__ void upsample_kernel(const float* __restrict__ src,  // [B,4,32,32]
                                float* __restrict__ dst)        // [B,4,64,64]
{
    int idx = blockIdx.x * blockDim.x + threadIdx.x;
    if (idx >= BATCH * 4 * 64 * 64) return;
    int x  = idx & 63;
    int y  = (idx >> 6) & 63;
    int bc = idx >> 12;             // b*4 + ch
    float sy = y * 0.5f - 0.25f;
    float sx = x * 0.5f - 0.25f;
    int y0 = (int)floorf(sy), x0 = (int)floorf(sx);
    float wy = sy - (float)y0, wx = sx - (float)x0;
    int y0c = min(max(y0,     0), 31), y1c = min(max(y0 + 1, 0), 31);
    int x0c = min(max(x0,     0), 31), x1c = min(max(x0 + 1, 0), 31);
    const float* s = src + bc * 1024;
    float v00 = s[y0c * 32 + x0c], v01 = s[y0c * 32 + x1c];
    float v10 = s[y1c * 32 + x0c], v11 = s[y1c * 32 + x1c];
    dst[idx] = (1.f - wy) * ((1.f - wx) * v00 + wx * v01)
             +        wy  * ((1.f - wx) * v10 + wx * v11);
}

// ---------------------------------------------------------------------------
// 1x1 convs: Q from second_frame (3ch), K from first_frame_aligned (3ch),
// V from upsampled features (4ch).  Layout [b][c][n], n fastest.
// ---------------------------------------------------------------------------
__global__ void qkv_kernel(const float* __restrict__ x2,  // [B,3,HW]
                           const float* __restrict__ x1,  // [B,3,HW]
                           const float* __restrict__ fu,  // [B,4,HW]
                           const float* __restrict__ Wq, const float* __restrict__ bq,
                           const float* __restrict__ Wk, const float* __restrict__ bk,
                           const float* __restrict__ Wv, const float* __restrict__ bv,
                           float* __restrict__ Q, float* __restrict__ K,
                           float* __restrict__ V)
{
    int idx = blockIdx.x * blockDim.x + threadIdx.x;
    if (idx >= BATCH * HD * HW) return;
    int n = idx & (HW - 1);
    int c = (idx >> 12) % HD;
    int b = idx / (HD * HW);

    const float* a2 = x2 + b * 3 * HW + n;
    const float* a1 = x1 + b * 3 * HW + n;
    const float* f  = fu + b * 4 * HW + n;

    float q = bq[c] + Wq[c*3+0]*a2[0] + Wq[c*3+1]*a2[HW] + Wq[c*3+2]*a2[2*HW];
    float k = bk[c] + Wk[c*3+0]*a1[0] + Wk[c*3+1]*a1[HW] + Wk[c*3+2]*a1[2*HW];
    float v = bv[c] + Wv[c*4+0]*f[0]  + Wv[c*4+1]*f[HW]  + Wv[c*4+2]*f[2*HW]
                    + Wv[c*4+3]*f[3*HW];
    Q[idx] = q;
    K[idx] = k;
    V[idx] = v;
}

__global__ void zero_kernel(float* __restrict__ p, int n)
{
    int idx = blockIdx.x * blockDim.x + threadIdx.x;
    if (idx < n) p[idx] = 0.0f;
}

// ---------------------------------------------------------------------------
// WMMA attention core.  One wave := 32 queries (2 subtiles of 16) of one
// (b, h).  Scans all keys in 16-wide tiles:
//   sim  = Q^T K  via  8x V_WMMA_F32_16X16X4_F32          (A=Q^T, B=K)
//   attn = relu((sim/sqrt(32) - t)/3000)                  (in D layout)
//   out^T += V * attn^T  via 8x WMMA per subtile          (A=V,  B=attn^T LDS)
// Then folds the per-head slice of the output projection (Wp) and
// atomically accumulates 4 channels into accum[b][4][n].
// ---------------------------------------------------------------------------
#define WAVES_PER_BLOCK 4
#define LDS_PER_WAVE (16*18 + 32 + 128)   // attn tile + t + Wp slice

__global__ __launch_bounds__(WAVES_PER_BLOCK * 32)
void CrossFrameAbsoluteAttn_attn_kernel(const float* __restrict__ Q,
                                        const float* __restrict__ K,
                                        const float* __restrict__ V,
                                        const float* __restrict__ Wt,
                                        const float* __restrict__ bt,
                                        const float* __restrict__ Wp,
                                        float* __restrict__ accum)
{
    const int lane = threadIdx.x & 31;
    const int wave = threadIdx.x >> 5;
    const int gw   = blockIdx.x * WAVES_PER_BLOCK + wave;  // 0 .. 2303
    const int qblk = gw & 127;                             // HW/32 strips
    const int bh   = gw >> 7;                              // 0 .. 17
    const int h    = bh % NHEADS;
    const int b    = bh / NHEADS;
    const int qbase = qblk * 32;

    const int half = lane >> 4;      // K-half selector for A/B fragments
    const int l16  = lane & 15;      // M (A) / N (B,C,D) index

    const float* Qb = Q + (size_t)(b * HD + h * EMBED) * HW;
    const float* Kb = K + (size_t)(b * HD + h * EMBED) * HW;
    const float* Vb = V + (size_t)(b * HD + h * EMBED) * HW;

    __shared__ float lds[WAVES_PER_BLOCK][LDS_PER_WAVE];
    float* attn_lds = &lds[wave][0];        // 16 rows (query) x stride 18 (key)
    float* t_lds    = &lds[wave][16*18];    // 32 per-query thresholds
    float* wp_lds   = &lds[wave][16*18+32]; // Wp[o][h*32 + d], 4x32

    // Stage this head's projection weights.
    for (int i = lane; i < 128; i += 32) {
        int o = i >> 5, d = i & 31;
        wp_lds[i] = Wp[o * HD + h * EMBED + d];
    }

    // Learned threshold t[q] = Wt . Q[:,q] + bt  (lane <-> query).
    {
        float acc = bt[0];
        for (int e = 0; e < EMBED; ++e)
            acc += Wt[e] * Qb[(size_t)e * HW + qbase + lane];
        t_lds[lane] = acc;
    }

    // Per-lane thresholds in D layout: row M = r + 8*half, per subtile.
    float thr[2][8];
    for (int s = 0; s < 2; ++s)
        for (int r = 0; r < 8; ++r)
            thr[s][r] = t_lds[s * 16 + r + 8 * half];

    // A fragments of Q^T (M=query, K=embed): lane holds K = 4i+2*half+{0,1}.
    v2f qa[2][8];
    for (int s = 0; s < 2; ++s)
        for (int i = 0; i < 8; ++i) {
            int d = 4 * i + 2 * half;
            int q = qbase + s * 16 + l16;
            qa[s][i].x = Qb[(size_t)d       * HW + q];
            qa[s][i].y = Qb[(size_t)(d + 1) * HW + q];
        }

    v8f acc_out[2][2] = {};   // [subtile][d-tile]: out^T, M=d_local, N=query

    const float scale  = 1.0f / (5.65685424949238f * 3000.0f); // 1/(sqrt(32)*T)
    const float tscale = 1.0f / 3000.0f;

    for (int kb = 0; kb < HW; kb += 16) {
        // B fragments of K (K=embed, N=key).
        v2f kf[8];
        for (int i = 0; i < 8; ++i) {
            int d = 4 * i + 2 * half;
            kf[i].x = Kb[(size_t)d       * HW + kb + l16];
            kf[i].y = Kb[(size_t)(d + 1) * HW + kb + l16];
        }
        // A fragments of V (M=d_local, K=key chunk): contiguous key pairs.
        v2f va[2][4];
        for (int dt = 0; dt < 2; ++dt)
            for (int c = 0; c < 4; ++c) {
                const float* p = Vb + (size_t)(dt * 16 + l16) * HW
                               + kb + c * 4 + 2 * half;
                va[dt][c].x = p[0];
                va[dt][c].y = p[1];
            }

        for (int s = 0; s < 2; ++s) {
            // sim tile: 16 queries x 16 keys, K-dim 32 in 8 steps of 4.
            v8f sim = {};
            for (int i = 0; i < 8; ++i)
                sim = __builtin_amdgcn_wmma_f32_16x16x4_f32(
                        false, qa[s][i], false, kf[i], (short)0, sim,
                        false, false);

            // attn = relu((sim/sqrt(d) - t)/T); stash [query][key] in LDS.
            for (int r = 0; r < 8; ++r) {
                float v = fmaxf(sim[r] * scale - thr[s][r] * tscale, 0.0f);
                attn_lds[(r + 8 * half) * 18 + l16] = v;
            }

            // out^T += V * attn^T   (B fragment from LDS, N=query).
            for (int c = 0; c < 4; ++c) {
                v2f bf;
                bf.x = attn_lds[l16 * 18 + c * 4 + 2 * half + 0];
                bf.y = attn_lds[l16 * 18 + c * 4 + 2 * half + 1];
                for (int dt = 0; dt < 2; ++dt)
                    acc_out[s][dt] = __builtin_amdgcn_wmma_f32_16x16x4_f32(
                            false, va[dt][c], false, bf, (short)0,
                            acc_out[s][dt], false, false);
            }
        }
    }

    // Fold output projection for this head: p[o] = sum_d Wp[o][h*32+d]*out[d][q]
    for (int s = 0; s < 2; ++s) {
        float p[4];
        for (int o = 0; o < 4; ++o) {
            float acc = 0.0f;
            for (int dt = 0; dt < 2; ++dt)
                for (int r = 0; r < 8; ++r)
                    acc += wp_lds[o * 32 + dt * 16 + r + 8 * half]
                         * acc_out[s][dt][r];
            p[o] = acc;
        }
        int q = qbase + s * 16 + l16;
        for (int o = 0; o < 4; ++o) {
            float v = p[o] + __shfl_xor(p[o], 16, 32);  // join the two d-halves
            if (half == 0)
                atomicAdd(&accum[(size_t)(b * 4 + o) * HW + q], v);
        }
    }
}

// ---------------------------------------------------------------------------
// 2x2 average (== linear resize 64->32, antialias=False) + bias + residual.
// d_out = [final_feat (8192) | out_feat_down (8192)]
// ---------------------------------------------------------------------------
__global__ void epilogue_kernel(const float* __restrict__ accum,  // [B,4,64,64]
                                const float* __restrict__ bp,
                                const float* __restrict__ feat2,  // [B,4,32,32]
                                float* __restrict__ out)
{
    int idx = blockIdx.x * blockDim.x + threadIdx.x;
    if (idx >= BATCH * 4 * 32 * 32) return;
    int x  = idx & 31;
    int y  = (idx >> 5) & 31;
    int bc = idx >> 10;
    int o  = bc & 3;
    const float* a = accum + (size_t)bc * HW;
    int Y = 2 * y, X = 2 * x;
    float d = 0.25f * (a[Y * 64 + X] + a[Y * 64 + X + 1]
                     + a[(Y + 1) * 64 + X] + a[(Y + 1) * 64 + X + 1]) + bp[o];
    out[idx]        = feat2[idx] + d;   // final_feat
    out[8192 + idx] = d;                // out_feat_down
}

// ---------------------------------------------------------------------------
extern "C" void kernel_launch(void* const* d_in, const int* in_sizes, int n_in,
                              void* d_out, int out_size, void* d_ws, size_t ws_size,
                              hipStream_t stream)
{
    const float* x2    = (const float*)d_in[0];   // second_frame
    const float* x1    = (const float*)d_in[1];   // first_frame_aligned
    const float* feat2 = (const float*)d_in[2];   // second_frame_feat
    const float* feat1 = (const float*)d_in[3];   // first_frame_feat_aligned
    const float* Wq    = (const float*)d_in[4];
    const float* bq    = (const float*)d_in[5];
    const float* Wk    = (const float*)d_in[6];
    const float* bk    = (const float*)d_in[7];
    const float* Wv    = (const float*)d_in[8];
    const float* bv    = (const float*)d_in[9];
    const float* Wp    = (const float*)d_in[10];
    const float* bp    = (const float*)d_in[11];
    const float* Wt    = (const float*)d_in[12];
    const float* bt    = (const float*)d_in[13];
    float* out = (float*)d_out;

    const size_t QSZ = (size_t)BATCH * HD * HW;   // 2,359,296 floats
    const size_t FSZ = (size_t)BATCH * 4 * HW;    //    32,768 floats
    const size_t need = (3 * QSZ + 2 * FSZ) * sizeof(float);
    if (ws_size < need) return;

    float* Q   = (float*)d_ws;
    float* K   = Q + QSZ;
    float* V   = K + QSZ;
    float* FU  = V + QSZ;     // upsampled features
    float* ACC = FU + FSZ;    // projected full-res output accumulator

    zero_kernel<<<(int)((FSZ + 255) / 256), 256, 0, stream>>>(ACC, (int)FSZ);
    upsample_kernel<<<(int)((FSZ + 255) / 256), 256, 0, stream>>>(feat1, FU);
    qkv_kernel<<<(int)((QSZ + 255) / 256), 256, 0, stream>>>(
        x2, x1, FU, Wq, bq, Wk, bk, Wv, bv, Q, K, V);

    // 2 batches * 9 heads * 128 query-strips = 2304 waves, 4 waves/block
    CrossFrameAbsoluteAttn_attn_kernel<<<576, WAVES_PER_BLOCK * 32, 0, stream>>>(
        Q, K, V, Wt, bt, Wp, ACC);

    epilogue_kernel<<<32, 256, 0, stream>>>(ACC, bp, feat2, out);
}